// FractalDimensionModule_652835029789
// MI455X (gfx1250) — compile-verified
//
#include <hip/hip_runtime.h>
#include <math.h>

// ---------------------------------------------------------------------------
// Fractal dimension via shared min/max pyramid + TDM tile load + WMMA box-sums
//
// Grid: 8x8 workgroups, 256 threads (8 wave32). Each WG:
//   1. TDM tensor_load_to_lds: 128x128 f32 tile (edge tiles shifted in-bounds)
//   2. Build M4 (32x32 grid of 4x4 px min/max), then sliding M8/M16/M32/M64
//      (power-of-two combines valid at EVERY 4-grid position, since window
//      origins are 4px aligned)
//   3. For s=4,8,16,32: window box-sums over D=max-min grids as a GEMM with
//      0/1 banded matrices: S = T_v * D * T_h^T, via v_wmma_f32_16x16x4_f32
//      (one wave per scale, K=32 in 8 chunks). s=64 is a direct lookup.
//   4. 5-point log-log regression -> fd, store 113x113 outputs.
// ---------------------------------------------------------------------------

typedef float v2f __attribute__((ext_vector_type(2)));
typedef float v8f __attribute__((ext_vector_type(8)));
typedef unsigned int v4u __attribute__((ext_vector_type(4)));
typedef int v8i __attribute__((ext_vector_type(8)));
typedef int v4i __attribute__((ext_vector_type(4)));

#define IMG   512
#define TILE  128
#define PITCH 33     // 32-wide grids padded to 33 to spread LDS banks
#define NOUT  113

// banded box-filter matrix entry: 1 if delta in {0, st, 2*st, ...} and < 16
__device__ __forceinline__ float band(int d, int st) {
  return (d >= 0 && d < 16 && ((d & (st - 1)) == 0)) ? 1.0f : 0.0f;
}

__device__ __forceinline__ void build_level(const float* pmin, const float* pmax,
                                            float* nmin, float* nmax, float* nd,
                                            int off, int lim, int tid) {
  for (int c = tid; c < 32 * 32; c += 256) {
    int p = c >> 5, q = c & 31;
    float mn = 0.0f, mx = 0.0f;  // zero-fill invalid cells (keeps 0*band clean)
    if (p <= lim && q <= lim) {
      mn = fminf(fminf(pmin[p * PITCH + q],        pmin[p * PITCH + q + off]),
                 fminf(pmin[(p + off) * PITCH + q], pmin[(p + off) * PITCH + q + off]));
      mx = fmaxf(fmaxf(pmax[p * PITCH + q],        pmax[p * PITCH + q + off]),
                 fmaxf(pmax[(p + off) * PITCH + q], pmax[(p + off) * PITCH + q + off]));
    }
    nmin[p * PITCH + q] = mn;
    nmax[p * PITCH + q] = mx;
    nd[p * PITCH + q]   = mx - mn;
  }
}

__global__ __launch_bounds__(256)
void FractalDimensionModule_652835029789_kernel(const float* __restrict__ X,
                                                float* __restrict__ out) {
  const int tid  = threadIdx.x;
  const int wave = tid >> 5;
  const int lane = tid & 31;
  const int hf   = lane >> 4;   // lane half (K split for WMMA A/B layouts)
  const int l15  = lane & 15;
  const int bx = blockIdx.x, by = blockIdx.y;
  // Edge tiles shifted fully in-bounds; window grid gets a +16 cell offset.
  const int offX = (bx == 7) ? 16 : 0;
  const int offY = (by == 7) ? 16 : 0;
  const int col0 = bx * 64 - offX * 4;
  const int row0 = by * 64 - offY * 4;

  // levels >= 8 overlay the pixel tile (dead after M4) to bound LDS use.
  __shared__ union {
    float tile[TILE * TILE];                       // 64 KB
    struct {
      float m8min[32 * PITCH],  m8max[32 * PITCH],  d8[32 * PITCH];
      float m16min[32 * PITCH], m16max[32 * PITCH], d16[32 * PITCH];
      float m32min[32 * PITCH], m32max[32 * PITCH], d32[32 * PITCH];
      float m64min[32 * PITCH], m64max[32 * PITCH], d64[32 * PITCH];
    } lv;
  } u;
  __shared__ float m4min[32 * PITCH], m4max[32 * PITCH], d4[32 * PITCH];
  __shared__ float Vbuf[4 * 16 * 32];   // per-scale GEMM intermediate
  __shared__ float Sbuf[4 * 256];       // per-scale window sums

  // ---- 1. Tile load: Tensor Data Mover (async DMA to LDS) ----------------
#if __has_builtin(__builtin_amdgcn_tensor_load_to_lds) && __has_builtin(__builtin_amdgcn_s_wait_tensorcnt)
  if (wave == 0) {
    unsigned long long ga =
        (unsigned long long)(size_t)(X + (size_t)row0 * IMG + col0);
    unsigned int lds = (unsigned int)(size_t)(&u.tile[0]);
    v4u g0;
    g0.x = 1u;                                   // count=1, user descriptor
    g0.y = lds;                                  // lds_addr [63:32]
    g0.z = (unsigned int)(ga & 0xFFFFFFFFu);     // global_addr [95:64]
    g0.w = (unsigned int)((ga >> 32) & 0x1FFFFFFu) | (2u << 30);  // type=2
    v8i g1;
    g1[0] = (int)(2u << 16);      // data_size=2 (4B), no multicast/pad/iter
    g1[1] = (int)(512u << 16);    // tensor_dim0 = 512
    g1[2] = (int)(512u << 16);    // tensor_dim1 = 512
    g1[3] = (int)(128u << 16);    // tile_dim0 = 128
    g1[4] = 128;                  // tile_dim1 = 128, tile_dim2 = 0
    g1[5] = 512;                  // tensor_dim0_stride = 512
    g1[6] = 0;
    g1[7] = 0;                    // tensor_dim1_stride = 0 (2D)
    v4i g2 = {0, 0, 0, 0}, g3 = {0, 0, 0, 0};
#if defined(__clang_major__) && (__clang_major__ >= 23)
    v8i g4 = {0, 0, 0, 0, 0, 0, 0, 0};
    __builtin_amdgcn_tensor_load_to_lds(g0, g1, g2, g3, g4, 0);
#else
    __builtin_amdgcn_tensor_load_to_lds(g0, g1, g2, g3, 0);
#endif
    __builtin_amdgcn_s_wait_tensorcnt(0);
  }
#else
  for (int idx = tid; idx < TILE * TILE; idx += 256) {
    int r = idx >> 7, c = idx & 127;
    u.tile[idx] = X[(size_t)(row0 + r) * IMG + (col0 + c)];
  }
#endif
  __syncthreads();

  // ---- 2. M4: 4x4-pixel block min/max over the whole tile ----------------
  for (int c = tid; c < 32 * 32; c += 256) {
    int cy = c >> 5, cx = c & 31;
    const float* p = &u.tile[(cy * 4) * TILE + cx * 4];
    float mn = p[0], mx = p[0];
#pragma unroll
    for (int r = 0; r < 4; ++r)
#pragma unroll
      for (int cc = 0; cc < 4; ++cc) {
        float v = p[r * TILE + cc];
        mn = fminf(mn, v);
        mx = fmaxf(mx, v);
      }
    m4min[cy * PITCH + cx] = mn;
    m4max[cy * PITCH + cx] = mx;
    d4[cy * PITCH + cx]    = mx - mn;
  }
  __syncthreads();  // tile dead past here; lv overlays it

  build_level(m4min, m4max, u.lv.m8min, u.lv.m8max, u.lv.d8, 1, 30, tid);
  __syncthreads();
  build_level(u.lv.m8min, u.lv.m8max, u.lv.m16min, u.lv.m16max, u.lv.d16, 2, 28, tid);
  __syncthreads();
  build_level(u.lv.m16min, u.lv.m16max, u.lv.m32min, u.lv.m32max, u.lv.d32, 4, 24, tid);
  __syncthreads();
  build_level(u.lv.m32min, u.lv.m32max, u.lv.m64min, u.lv.m64max, u.lv.d64, 8, 16, tid);
  __syncthreads();

  // ---- 3. Window box-sums: banded GEMM on the matrix cores ---------------
  // NOTE: no pointer arrays here (ld.lld can't emit addrspacecast constant
  // initializers for LDS); select the per-scale D grid via runtime selects.
#if __has_builtin(__builtin_amdgcn_wmma_f32_16x16x4_f32)
  if (wave < 4) {                 // wave sc handles scale s = 4 << sc
    const int sc = wave;
    const int st = 1 << sc;       // grid-cell stride between blocks
    const float* Dl = (sc == 0) ? (const float*)d4
                    : (sc == 1) ? (const float*)u.lv.d8
                    : (sc == 2) ? (const float*)u.lv.d16
                                : (const float*)u.lv.d32;
    float* Vst = &Vbuf[sc * 16 * 32];
    // Step A: V[m][q] = sum_k band(k-m-offY) * D[k][q]   (vertical box)
#pragma unroll
    for (int qt = 0; qt < 2; ++qt) {
      const int qb = qt * 16;
      v8f acc = {};
#pragma unroll
      for (int k0 = 0; k0 < 32; k0 += 4) {
        int ks = k0 + 2 * hf;     // A/B layout: lane halves take K pairs
        v2f a, b;
        a.x = band(ks + 0 - l15 - offY, st);
        a.y = band(ks + 1 - l15 - offY, st);
        b.x = Dl[(ks + 0) * PITCH + qb + l15];
        b.y = Dl[(ks + 1) * PITCH + qb + l15];
        acc = __builtin_amdgcn_wmma_f32_16x16x4_f32(false, a, false, b,
                                                    0, acc, false, false);
      }
#pragma unroll
      for (int v = 0; v < 8; ++v)
        Vst[(v + 8 * hf) * 32 + qb + l15] = acc[v];   // C layout -> LDS
    }
    // Step B: S[m][n] = sum_q V[m][q] * band(q-n-offX)  (horizontal box)
    v8f acc = {};
#pragma unroll
    for (int k0 = 0; k0 < 32; k0 += 4) {
      int ks = k0 + 2 * hf;
      v2f a, b;
      a.x = Vst[l15 * 32 + ks + 0];
      a.y = Vst[l15 * 32 + ks + 1];
      b.x = band(ks + 0 - l15 - offX, st);
      b.y = band(ks + 1 - l15 - offX, st);
      acc = __builtin_amdgcn_wmma_f32_16x16x4_f32(false, a, false, b,
                                                  0, acc, false, false);
    }
#pragma unroll
    for (int v = 0; v < 8; ++v)
      Sbuf[sc * 256 + (v + 8 * hf) * 16 + l15] = acc[v];
  }
#else
  {  // VALU fallback: direct strided sums
    const int i = tid >> 4, j = tid & 15;
    if (by * 16 + i < NOUT && bx * 16 + j < NOUT) {
      for (int sc = 0; sc < 4; ++sc) {
        int st = 1 << sc, terms = 16 >> sc;
        const float* Dl = (sc == 0) ? (const float*)d4
                        : (sc == 1) ? (const float*)u.lv.d8
                        : (sc == 2) ? (const float*)u.lv.d16
                                    : (const float*)u.lv.d32;
        float s = 0.0f;
        for (int a = 0; a < terms; ++a)
          for (int b = 0; b < terms; ++b)
            s += Dl[(offY + i + a * st) * PITCH + offX + j + b * st];
        Sbuf[sc * 256 + i * 16 + j] = s;
      }
    } else {
      for (int sc = 0; sc < 4; ++sc) Sbuf[sc * 256 + (tid & 255)] = 1.0f;
    }
  }
#endif
  __syncthreads();

  // ---- 4. log-log regression and store -----------------------------------
  const int i = tid >> 4, j = tid & 15;
  const int gr = by * 16 + i, gc = bx * 16 + j;
  if (gr < NOUT && gc < NOUT) {
    float db4  = Sbuf[0 * 256 + i * 16 + j];
    float db8  = Sbuf[1 * 256 + i * 16 + j];
    float db16 = Sbuf[2 * 256 + i * 16 + j];
    float db32 = Sbuf[3 * 256 + i * 16 + j];
    float db64 = u.lv.d64[(offY + i) * PITCH + offX + j];
    const float L64 = 4.1588830833596715f, L32 = 3.4657359027997265f,
                L16 = 2.7725887222397810f, L8  = 2.0794415416798357f,
                L4  = 1.3862943611198906f;
    const float inv = -1.0f / (L64 * L64 + L32 * L32 + L16 * L16 +
                               L8 * L8 + L4 * L4);
    float fd = inv * (L64 * __logf(db64) + L32 * __logf(db32) +
                      L16 * __logf(db16) + L8 * __logf(db8) +
                      L4 * __logf(db4));
    out[gr * NOUT + gc] = fd;
  }
}

extern "C" void kernel_launch(void* const* d_in, const int* in_sizes, int n_in,
                              void* d_out, int out_size, void* d_ws, size_t ws_size,
                              hipStream_t stream) {
  (void)in_sizes; (void)n_in; (void)d_ws; (void)ws_size; (void)out_size;
  const float* X = (const float*)d_in[0];
  float* out = (float*)d_out;
  dim3 grid(8, 8, 1), block(256, 1, 1);
  FractalDimensionModule_652835029789_kernel<<<grid, block, 0, stream>>>(X, out);
}